// Decoder_9234179686937
// MI455X (gfx1250) — compile-verified
//
#include <hip/hip_runtime.h>
#include <hip/hip_bf16.h>
#include <math.h>

#define NEG_INF (-1.0e15f)

typedef float v2f __attribute__((ext_vector_type(2)));
typedef float v8f __attribute__((ext_vector_type(8)));

// D(16x16,f32) = A(16x4,f32) x B(4x16,f32) + C
// 8-arg form: (neg_a, A, neg_b, B, c_mod, C, reuse_a, reuse_b)
__device__ __forceinline__ v8f wmma_f32_16x16x4(v2f a, v2f b, v8f c) {
    return __builtin_amdgcn_wmma_f32_16x16x4_f32(false, a, false, b, (short)0, c,
                                                 false, false);
}

// One workgroup (256 threads = 8 wave32) per batch element.
// B=256, N=1024, D=128, H=8, HD=16.
__global__ __launch_bounds__(256) void decoder_step_kernel(
    const float* __restrict__ h_g,     // [B,128]
    const float* __restrict__ first,   // [B,128]
    const float* __restrict__ last,    // [B,128]
    const float* __restrict__ context, // [B,2]
    const float* __restrict__ K,       // [B,8,1024,16]
    const float* __restrict__ V,       // [B,8,1024,16]
    const float* __restrict__ K_lg,    // [B,1024,128]
    const float* __restrict__ Wq,      // [128,386]
    const float* __restrict__ bq,      // [128]
    const float* __restrict__ Wo,      // [128,128]
    const float* __restrict__ bo,      // [128]
    const int*   __restrict__ mask,    // [B,1024]
    float* __restrict__ out)           // [512]: vertexes(256) then probs(256)
{
    const int b    = blockIdx.x;
    const int tid  = threadIdx.x;
    const int lane = tid & 31;
    const int wave = tid >> 5;     // == head index for phases 1-2
    const int half = lane >> 4;    // 0: lanes 0-15 (k=0,1) | 1: lanes 16-31 (k=2,3)
    const int lk   = half * 2;     // k-offset within a 4-chunk for this lane half
    const int ln   = lane & 15;    // column (n) index within a 16-tile

    __shared__ float lds_Q[128];
    __shared__ float lds_att[8 * 1024];   // per-head scores -> exp weights
    __shared__ float lds_u[128];
    __shared__ float lds_uo[128];
    __shared__ float lds_logits[1024];
    __shared__ int   lds_mask[1024];
    __shared__ float s_red[256];
    __shared__ int   s_ridx[256];

    // ---------------- Phase 0a: stage mask row in LDS ----------------
    {
        const int* mrow = mask + (size_t)b * 1024;
        #pragma unroll
        for (int i = 0; i < 4; ++i) lds_mask[tid + 256 * i] = mrow[tid + 256 * i];
    }

    // ---------------- Phase 0b: Q = h_c @ Wq^T + bq ----------------
    if (tid < 128) {
        const float* wrow = Wq + (size_t)tid * 386;
        float acc = bq[tid];
        const float* hg = h_g   + (size_t)b * 128;
        const float* fi = first + (size_t)b * 128;
        const float* la = last  + (size_t)b * 128;
        #pragma unroll 4
        for (int j = 0; j < 128; ++j) {
            acc += hg[j] * wrow[j];
            acc += fi[j] * wrow[128 + j];
            acc += la[j] * wrow[256 + j];
        }
        acc += context[b * 2 + 0] * wrow[384];
        acc += context[b * 2 + 1] * wrow[385];
        lds_Q[tid] = acc;
    }
    __syncthreads();

    // ---------------- Phase 1: scores = Q . K / 4, mask ----------------
    // Wave w handles head w. D[m][n] = sum_k q[k] * K[n0+n][k]; A rows all = q.
    {
        const float* Kb = K + ((size_t)(b * 8 + wave) * 1024) * 16;
        v2f qa[4];
        #pragma unroll
        for (int j = 0; j < 4; ++j) {
            qa[j].x = lds_Q[wave * 16 + 4 * j + lk];
            qa[j].y = lds_Q[wave * 16 + 4 * j + lk + 1];
        }
        for (int t = 0; t < 64; ++t) {
            const int n0 = t * 16;
            const float* krow = Kb + (size_t)(n0 + ln) * 16;
            v8f c = {};
            #pragma unroll
            for (int j = 0; j < 4; ++j) {
                v2f bb = *(const v2f*)(krow + 4 * j + lk);   // b64 load
                c = wmma_f32_16x16x4(qa[j], bb, c);
            }
            const float sc = c[0] * 0.25f;                   // / sqrt(16)
            const int   mk = lds_mask[n0 + ln];
            if (lane < 16)
                lds_att[wave * 1024 + n0 + lane] = (mk == 0) ? NEG_INF : sc;
        }
    }

    // ---------------- per-head softmax (same wave owns its region) -------
    float inv_sum;
    {
        float lmax = -3.0e38f;
        #pragma unroll 4
        for (int i = 0; i < 32; ++i)
            lmax = fmaxf(lmax, lds_att[wave * 1024 + i * 32 + lane]);
        for (int off = 16; off > 0; off >>= 1)
            lmax = fmaxf(lmax, __shfl_xor(lmax, off, 32));
        float lsum = 0.0f;
        #pragma unroll 4
        for (int i = 0; i < 32; ++i) {
            const int idx = wave * 1024 + i * 32 + lane;
            float e = __expf(lds_att[idx] - lmax);
            lds_att[idx] = e;                  // keep unnormalized weights
            lsum += e;
        }
        for (int off = 16; off > 0; off >>= 1)
            lsum += __shfl_xor(lsum, off, 32);
        inv_sum = 1.0f / lsum;
    }

    // ---------------- Phase 2: u = (attn . V) * inv_sum ----------------
    // D[m][d] = sum_n w[n] * V[n][d]; A rows all = w-chunk.
    // 4 independent accumulators break the WMMA->WMMA RAW hazard chain.
    {
        const float* Vb = V + ((size_t)(b * 8 + wave) * 1024) * 16;
        v8f cu0 = {}, cu1 = {}, cu2 = {}, cu3 = {};
        for (int kk = 0; kk < 256; kk += 4) {
            v2f a0, a1, a2, a3, b0, b1, b2, b3;
            const int n0 = kk * 4;
            a0.x = lds_att[wave * 1024 + n0 +  0 + lk];
            a0.y = lds_att[wave * 1024 + n0 +  1 + lk];
            a1.x = lds_att[wave * 1024 + n0 +  4 + lk];
            a1.y = lds_att[wave * 1024 + n0 +  5 + lk];
            a2.x = lds_att[wave * 1024 + n0 +  8 + lk];
            a2.y = lds_att[wave * 1024 + n0 +  9 + lk];
            a3.x = lds_att[wave * 1024 + n0 + 12 + lk];
            a3.y = lds_att[wave * 1024 + n0 + 13 + lk];
            b0.x = Vb[(size_t)(n0 +  0 + lk) * 16 + ln];
            b0.y = Vb[(size_t)(n0 +  1 + lk) * 16 + ln];
            b1.x = Vb[(size_t)(n0 +  4 + lk) * 16 + ln];
            b1.y = Vb[(size_t)(n0 +  5 + lk) * 16 + ln];
            b2.x = Vb[(size_t)(n0 +  8 + lk) * 16 + ln];
            b2.y = Vb[(size_t)(n0 +  9 + lk) * 16 + ln];
            b3.x = Vb[(size_t)(n0 + 12 + lk) * 16 + ln];
            b3.y = Vb[(size_t)(n0 + 13 + lk) * 16 + ln];
            cu0 = wmma_f32_16x16x4(a0, b0, cu0);
            cu1 = wmma_f32_16x16x4(a1, b1, cu1);
            cu2 = wmma_f32_16x16x4(a2, b2, cu2);
            cu3 = wmma_f32_16x16x4(a3, b3, cu3);
        }
        const float ures = (cu0[0] + cu1[0]) + (cu2[0] + cu3[0]);
        if (lane < 16) lds_u[wave * 16 + lane] = ures * inv_sum;
    }
    __syncthreads();

    // ---------------- Phase 3: u_o = u @ Wo^T + bo ----------------
    if (tid < 128) {
        const float* wrow = Wo + (size_t)tid * 128;
        float acc = bo[tid];
        #pragma unroll 4
        for (int e = 0; e < 128; ++e) acc += wrow[e] * lds_u[e];
        lds_uo[tid] = acc;
    }
    __syncthreads();

    // ---------------- Phase 4: logits = 10*tanh(u_o . K_lg / sqrt(128)) --
    {
        const float* Kl = K_lg + (size_t)b * 1024 * 128;
        v2f ua[32];
        #pragma unroll
        for (int j = 0; j < 32; ++j) {
            ua[j].x = lds_uo[4 * j + lk];
            ua[j].y = lds_uo[4 * j + lk + 1];
        }
        for (int ti = wave; ti < 64; ti += 8) {
            const int n0 = ti * 16;
            const float* row = Kl + (size_t)(n0 + ln) * 128;
            __builtin_prefetch(row + (size_t)128 * 128, 0, 3);  // next tile for this wave
            v8f c0 = {}, c1 = {};
            #pragma unroll 8
            for (int j = 0; j < 32; j += 2) {
                v2f bb0 = *(const v2f*)(row + 4 * j + lk);          // b64 load
                v2f bb1 = *(const v2f*)(row + 4 * (j + 1) + lk);    // b64 load
                c0 = wmma_f32_16x16x4(ua[j], bb0, c0);
                c1 = wmma_f32_16x16x4(ua[j + 1], bb1, c1);
            }
            float lg = (c0[0] + c1[0]) * 0.08838834764831844f;      // 1/sqrt(128)
            lg = 10.0f * tanhf(lg);
            const int mk = lds_mask[n0 + ln];
            if (lane < 16)
                lds_logits[n0 + lane] = (mk == 0) ? NEG_INF : lg;
        }
    }
    __syncthreads();

    // ---------------- Phase 5: softmax + argmax over 1024 ----------------
    {
        float bmax = -3.0e38f; int bidx = 0;
        #pragma unroll
        for (int i = 0; i < 4; ++i) {
            const int n = tid + 256 * i;
            const float v = lds_logits[n];
            if (v > bmax) { bmax = v; bidx = n; }
        }
        s_red[tid] = bmax; s_ridx[tid] = bidx;
        __syncthreads();
        for (int s = 128; s > 0; s >>= 1) {
            if (tid < s) {
                if (s_red[tid + s] > s_red[tid]) {
                    s_red[tid]  = s_red[tid + s];
                    s_ridx[tid] = s_ridx[tid + s];
                }
            }
            __syncthreads();
        }
        const float gmax = s_red[0];
        const int   gidx = s_ridx[0];
        __syncthreads();

        float lsum = 0.0f;
        #pragma unroll
        for (int i = 0; i < 4; ++i)
            lsum += __expf(lds_logits[tid + 256 * i] - gmax);
        s_red[tid] = lsum;
        __syncthreads();
        for (int s = 128; s > 0; s >>= 1) {
            if (tid < s) s_red[tid] += s_red[tid + s];
            __syncthreads();
        }
        if (tid == 0) {
            out[b]       = (float)gidx;          // vertexes (greedy argmax)
            out[256 + b] = 1.0f / s_red[0];      // p[argmax] = 1 / sum(exp(l-max))
        }
    }
}

extern "C" void kernel_launch(void* const* d_in, const int* in_sizes, int n_in,
                              void* d_out, int out_size, void* d_ws, size_t ws_size,
                              hipStream_t stream) {
    (void)in_sizes; (void)n_in; (void)d_ws; (void)ws_size; (void)out_size;
    // setup_inputs order:
    // 0:x 1:h_g 2:first 3:last 4:context 5:K 6:V 7:K_lg 8:Wq 9:bq 10:Wo 11:bo 12:mask 13:t
    const float* h_g     = (const float*)d_in[1];
    const float* first   = (const float*)d_in[2];
    const float* last    = (const float*)d_in[3];
    const float* context = (const float*)d_in[4];
    const float* K       = (const float*)d_in[5];
    const float* V       = (const float*)d_in[6];
    const float* K_lg    = (const float*)d_in[7];
    const float* Wq      = (const float*)d_in[8];
    const float* bq      = (const float*)d_in[9];
    const float* Wo      = (const float*)d_in[10];
    const float* bo      = (const float*)d_in[11];
    const int*   mask    = (const int*)d_in[12];
    float* out = (float*)d_out;

    decoder_step_kernel<<<256, 256, 0, stream>>>(h_g, first, last, context, K, V,
                                                 K_lg, Wq, bq, Wo, bo, mask, out);
}